// RTAttention_6717328851665
// MI455X (gfx1250) — compile-verified
//
#include <hip/hip_runtime.h>
#include <hip/hip_bf16.h>

// RT attention, fused. B=2, N=384, D_NODE=256, D_EDGE=32, D_HID=128, H=8, DH=16.
// Outputs: out [2,384,256] then attn [2,8,384,384], concatenated in d_out.

typedef float v2f __attribute__((ext_vector_type(2)));
typedef float v8f __attribute__((ext_vector_type(8)));
typedef int   v4i __attribute__((ext_vector_type(4)));
typedef __attribute__((address_space(1))) v4i g_v4i;   // global int4
typedef __attribute__((address_space(3))) v4i l_v4i;   // LDS int4

#define BATCH   2
#define NN      384
#define DNODE   256
#define DEDGE   32
#define DHID    128
#define NH      8
#define DH      16
#define QKV     384          // 3*DHID
#define RT_SCALE 0.08838834764831845f   // 1/sqrt(128)

// ---------------------------------------------------------------------------
// Generic f32 WMMA GEMM: C[M,N] = A[M,K] @ B[K,N] (+ bias[N]), row-major.
// One wave per 16x16 output tile; 8 waves per 256-thread block.
// K multiple of 4; M,N multiples of 16; grid covers tiles exactly.
// ---------------------------------------------------------------------------
__global__ __launch_bounds__(256) void wmma_gemm_f32(
    const float* __restrict__ A, const float* __restrict__ B,
    const float* __restrict__ bias, float* __restrict__ C,
    int M, int Nn, int K, int tilesN) {
  int wave = blockIdx.x * 8 + (threadIdx.x >> 5);
  int tm = wave / tilesN;
  int tn = wave % tilesN;
  int l    = threadIdx.x & 31;
  int half = l >> 4;          // 0: lanes 0-15, 1: lanes 16-31
  int lm   = l & 15;

  v8f c = {};
  const float* Arow = A + (size_t)(tm * 16 + lm) * K;
  for (int ks = 0; ks < K; ks += 4) {
    int kk = ks + 2 * half;                       // kk is even -> 8B aligned
    v2f a = *(const v2f*)(Arow + kk);
    v2f b; b.x = B[(size_t)kk * Nn + tn * 16 + lm];
           b.y = B[(size_t)(kk + 1) * Nn + tn * 16 + lm];
    c = __builtin_amdgcn_wmma_f32_16x16x4_f32(false, a, false, b,
                                              (short)0, c, false, false);
  }
  int n = tn * 16 + lm;
  float bv = bias ? bias[n] : 0.0f;
  for (int r = 0; r < 8; ++r) {
    int m = tm * 16 + r + 8 * half;
    C[(size_t)m * Nn + n] = c[r] + bv;
  }
}

// ---------------------------------------------------------------------------
// Fused per-(b,i) attention row. Block = 256 threads = 8 waves; wave h = head h.
// LDS: E tile 384x32 (48KB) | W_qkv_edge 32x384 (48KB) | dots 8x384 (12KB).
// ---------------------------------------------------------------------------
__global__ __launch_bounds__(256) void rt_attn_fused(
    const float* __restrict__ edge,      // [B,N,N,DEDGE]
    const float* __restrict__ qkvn,      // [B,N,QKV]  (precomputed node qkv)
    const float* __restrict__ Wedge,     // [DEDGE,QKV]
    float* __restrict__ attn_out,        // [B,H,N,N]
    float* __restrict__ out_hid) {       // [B,N,DHID]
  extern __shared__ float lds[];
  float* Elds = lds;               // 12288 floats
  float* Wlds = lds + 12288;       // 12288 floats
  float* Dlds = lds + 24576;       // 3072 floats (per-head dots/probs)

  const int bi = blockIdx.x;       // 0..767
  const int b  = bi / NN;
  const int i  = bi % NN;
  const int t  = threadIdx.x;

  // --- Stage edge row tile (384x32) and full W_qkv_edge (32x384) into LDS ---
  const float4* Eg = (const float4*)(edge + (size_t)(b * NN + i) * NN * DEDGE);
  const float4* Wg = (const float4*)Wedge;
  float4* E4 = (float4*)Elds;
  float4* W4 = (float4*)Wlds;
#if __has_builtin(__builtin_amdgcn_global_load_async_to_lds_b128)
  // CDNA5 async-copy path: GLOBAL_LOAD_ASYNC_TO_LDS_B128 (ASYNCcnt-tracked,
  // no VGPR round trip). Signature: (v4i AS1*, v4i AS3*, imm offset, imm cpol).
  for (int idx = t; idx < 3072; idx += 256) {
    __builtin_amdgcn_global_load_async_to_lds_b128(
        (g_v4i*)(Eg + idx), (l_v4i*)(E4 + idx), 0, 0);
    __builtin_amdgcn_global_load_async_to_lds_b128(
        (g_v4i*)(Wg + idx), (l_v4i*)(W4 + idx), 0, 0);
  }
#if __has_builtin(__builtin_amdgcn_s_wait_asynccnt)
  __builtin_amdgcn_s_wait_asynccnt(0);
#else
  asm volatile("s_wait_asynccnt 0" ::: "memory");
#endif
#else
  for (int idx = t; idx < 3072; idx += 256) E4[idx] = Eg[idx];
  for (int idx = t; idx < 3072; idx += 256) W4[idx] = Wg[idx];
#endif
  __syncthreads();

  const int h    = t >> 5;         // head == wave id
  const int l    = t & 31;
  const int half = l >> 4;
  const int lm   = l & 15;
  const int d    = lm;             // dh index carried by this lane

  float* dots = Dlds + h * NN;
  const float qn = qkvn[(size_t)(b * NN + i) * QKV + h * 48 + d];

  // Loop-invariant B fragments for this head: q/k/v columns of W_qkv_edge.
  // B 4x16 layout (mirrors A 16x4): VGPR0 = row K (lanes 0-15) / K+2 (16-31),
  // VGPR1 = K+1 / K+3.
  v2f bq[8], bk[8], bv[8];
  for (int ks = 0; ks < 8; ++ks) {
    int kk = ks * 4 + 2 * half;
    const float* B0 = Wlds + kk * QKV + h * 48 + lm;
    const float* B1 = Wlds + (kk + 1) * QKV + h * 48 + lm;
    bq[ks].x = B0[0];  bq[ks].y = B1[0];
    bk[ks].x = B0[16]; bk[ks].y = B1[16];
    bv[ks].x = B0[32]; bv[ks].y = B1[32];
  }

  // ---- Pass 1: raw dots for all 384 j (WMMA edge-QKV, Q & K components) ----
  for (int jt = 0; jt < 24; ++jt) {
    v8f cq = {}; v8f ck = {};
    const float* Arow = Elds + (jt * 16 + lm) * DEDGE;
    for (int ks = 0; ks < 8; ++ks) {
      int kk = ks * 4 + 2 * half;                 // even -> 8B-aligned b64 load
      v2f a = *(const v2f*)(Arow + kk);
      cq = __builtin_amdgcn_wmma_f32_16x16x4_f32(false, a, false, bq[ks],
                                                 (short)0, cq, false, false);
      ck = __builtin_amdgcn_wmma_f32_16x16x4_f32(false, a, false, bk[ks],
                                                 (short)0, ck, false, false);
    }
    // C/D layout: VGPR r holds (j_local = r + 8*half, d = lm)
    for (int r = 0; r < 8; ++r) {
      int j = jt * 16 + r + 8 * half;
      float kn = qkvn[(size_t)(b * NN + j) * QKV + h * 48 + 16 + d];
      float p  = (qn + cq[r]) * (kn + ck[r]);
      for (int off = 1; off < 16; off <<= 1) p += __shfl_xor(p, off, 32);
      if (lm == 0) dots[j] = p * RT_SCALE;        // lanes 0 and 16 write their j
    }
  }

  // ---- Softmax over j (wave-local), write attn, keep probs in LDS ----
  float vals[12];
  float mx = -1e30f;
  for (int tt = 0; tt < 12; ++tt) {
    vals[tt] = dots[l + 32 * tt];
    mx = fmaxf(mx, vals[tt]);
  }
  for (int off = 1; off < 32; off <<= 1) mx = fmaxf(mx, __shfl_xor(mx, off, 32));
  float sum = 0.0f;
  for (int tt = 0; tt < 12; ++tt) {
    vals[tt] = __expf(vals[tt] - mx);
    sum += vals[tt];
  }
  for (int off = 1; off < 32; off <<= 1) sum += __shfl_xor(sum, off, 32);
  float inv = 1.0f / sum;
  float* attn_row = attn_out + (size_t)((b * NH + h) * NN + i) * NN;
  for (int tt = 0; tt < 12; ++tt) {
    float p = vals[tt] * inv;
    dots[l + 32 * tt] = p;
    attn_row[l + 32 * tt] = p;
  }

  // ---- Pass 2: out[d] = sum_j p[j] * (v_n[j,d] + v_e[i,j,d]); V via WMMA ----
  float o = 0.0f;
  for (int jt = 0; jt < 24; ++jt) {
    v8f cv = {};
    const float* Arow = Elds + (jt * 16 + lm) * DEDGE;
    for (int ks = 0; ks < 8; ++ks) {
      int kk = ks * 4 + 2 * half;
      v2f a = *(const v2f*)(Arow + kk);
      cv = __builtin_amdgcn_wmma_f32_16x16x4_f32(false, a, false, bv[ks],
                                                 (short)0, cv, false, false);
    }
    for (int r = 0; r < 8; ++r) {
      int j = jt * 16 + r + 8 * half;
      float vn = qkvn[(size_t)(b * NN + j) * QKV + h * 48 + 32 + d];
      o += dots[j] * (vn + cv[r]);
    }
  }
  o += __shfl_xor(o, 16, 32);      // combine the two half-wave j partitions
  if (half == 0)
    out_hid[(size_t)(b * NN + i) * DHID + h * DH + d] = o;
}

// ---------------------------------------------------------------------------
extern "C" void kernel_launch(void* const* d_in, const int* in_sizes, int n_in,
                              void* d_out, int out_size, void* d_ws, size_t ws_size,
                              hipStream_t stream) {
  (void)in_sizes; (void)n_in; (void)out_size; (void)ws_size;
  const float* node  = (const float*)d_in[0];   // [2,384,256]
  const float* edge  = (const float*)d_in[1];   // [2,384,384,32]
  // d_in[2] = mask (unused by the module)
  const float* Wqn   = (const float*)d_in[3];   // [256,384]
  const float* Wqe   = (const float*)d_in[4];   // [32,384]
  const float* Wproj = (const float*)d_in[5];   // [128,256]
  const float* bproj = (const float*)d_in[6];   // [256]

  float* out_main = (float*)d_out;                                // 2*384*256
  float* attn_out = (float*)d_out + (size_t)BATCH * NN * DNODE;   // 2*8*384*384

  float* qkvn    = (float*)d_ws;                                  // 2*384*384
  float* out_hid = qkvn + (size_t)BATCH * NN * QKV;               // 2*384*128

  // K1: qkv_n = node @ W_qkv_node   (M=768, N=384, K=256) -> 1152 tiles / 8
  wmma_gemm_f32<<<144, 256, 0, stream>>>(node, Wqn, nullptr, qkvn,
                                         BATCH * NN, QKV, DNODE, QKV / 16);

  // K2: fused edge-QKV + attention + attn output + per-head context
  rt_attn_fused<<<BATCH * NN, 256, 110592, stream>>>(edge, qkvn, Wqe,
                                                     attn_out, out_hid);

  // K3: out = out_hid @ W_proj + b_proj  (M=768, N=256, K=128) -> 768 tiles / 8
  wmma_gemm_f32<<<96, 256, 0, stream>>>(out_hid, Wproj, bproj, out_main,
                                        BATCH * NN, DNODE, DHID, DNODE / 16);
}